// LookupNetwork_83726092469041
// MI455X (gfx1250) — compile-verified
//
#include <hip/hip_runtime.h>

// Embedding-table gather with -1 sentinel -> zero row.
// out[row, 0:128] = (idx[row] >= 0) ? table[idx[row], 0:128] : 0
// Memory-bound: ~420 MB NT stores to HBM; table (51 MB) stays resident in 192 MB L2.

typedef float v4f __attribute__((ext_vector_type(4)));

#define THREADS        256                      // 8 wave32's per block
#define WAVES          (THREADS / 32)
#define ROWS_PER_BLOCK 64
#define ROWS_PER_WAVE  (ROWS_PER_BLOCK / WAVES) // 8
#define ROW_BYTES      512                      // 128 floats * 4 B

__global__ __launch_bounds__(THREADS)
void emb_gather_kernel(const int* __restrict__ idx,
                       const float* __restrict__ table,
                       float* __restrict__ out,
                       int n_rows) {
  __shared__ int s_idx[ROWS_PER_BLOCK];

  const int tid  = threadIdx.x;
  const int lane = tid & 31;
  const int wave = tid >> 5;
  const long long blockRow0 = (long long)blockIdx.x * ROWS_PER_BLOCK;

#if defined(__gfx1250__)
  // Stage this block's 64 indices into LDS via the CDNA5 async DMA path:
  // wave 0, each lane moves 8 bytes (2 indices) global -> LDS (ASYNCcnt).
  if (wave == 0) {
    long long g = blockRow0 + 2 * lane;
    long long gmax = (long long)n_rows - 2;
    if (g > gmax) g = gmax;          // clamp keeps the b64 read in-bounds;
    if (g < 0)    g = 0;             // clamped slots belong to rows we skip
    unsigned voff  = (unsigned)(g * sizeof(int));                     // byte offset from SGPR base
    unsigned lds_b = (unsigned)(unsigned long long)&s_idx[2 * lane];  // low 32 bits = LDS byte addr
    asm volatile("global_load_async_to_lds_b64 %0, %1, %2"
                 :
                 : "v"(lds_b), "v"(voff), "s"(idx)
                 : "memory");
    asm volatile("s_wait_asynccnt 0" ::: "memory");
  }
  __syncthreads();
#else
  if (tid < ROWS_PER_BLOCK) {
    long long g = blockRow0 + tid;
    s_idx[tid] = (g < (long long)n_rows) ? idx[g] : -1;
  }
  __syncthreads();
#endif

  const char* __restrict__ tabB = (const char*)table;
  char* __restrict__       outB = (char*)out;
  const unsigned laneB = (unsigned)lane * 16u;           // lane's float4 slot in a row

  if (blockRow0 + ROWS_PER_BLOCK <= (long long)n_rows) {
    // Fast path: full block, no per-row bounds checks.
    // 32-bit offsets fit: table span 51.2 MB, output span 419 MB < 4 GB,
    // so loads/stores select the saddr + zext(u32 voffset) global form.
#pragma unroll
    for (int r = 0; r < ROWS_PER_WAVE; ++r) {
      const int localRow = wave + r * WAVES;             // 8 adjacent rows per step
      const unsigned row = (unsigned)blockRow0 + (unsigned)localRow;
      const int id = s_idx[localRow];                    // wave-uniform
      v4f v = {0.0f, 0.0f, 0.0f, 0.0f};
      if (id >= 0) {                                     // uniform branch, no divergence
        const unsigned toff = (unsigned)id * ROW_BYTES + laneB;
        v = *(const v4f*)(tabB + (size_t)toff);          // temporal: table is L2-resident
      }
      const unsigned ooff = row * ROW_BYTES + laneB;
      // Non-temporal: output is streamed once; don't evict the table from L2.
      __builtin_nontemporal_store(v, (v4f*)(outB + (size_t)ooff));
    }
  } else {
    // Tail path: guarded per row.
#pragma unroll
    for (int r = 0; r < ROWS_PER_WAVE; ++r) {
      const int localRow = wave + r * WAVES;
      const long long row = blockRow0 + localRow;
      if (row < (long long)n_rows) {
        const int id = s_idx[localRow];
        v4f v = {0.0f, 0.0f, 0.0f, 0.0f};
        if (id >= 0) {
          const unsigned toff = (unsigned)id * ROW_BYTES + laneB;
          v = *(const v4f*)(tabB + (size_t)toff);
        }
        const unsigned ooff = (unsigned)row * ROW_BYTES + laneB;
        __builtin_nontemporal_store(v, (v4f*)(outB + (size_t)ooff));
      }
    }
  }
}

extern "C" void kernel_launch(void* const* d_in, const int* in_sizes, int n_in,
                              void* d_out, int out_size, void* d_ws, size_t ws_size,
                              hipStream_t stream) {
  (void)n_in; (void)out_size; (void)d_ws; (void)ws_size;

  const int*   idx   = (const int*)d_in[0];    // input_batch, flat [BATCH*HIST]
  const float* table = (const float*)d_in[1];  // [VOCAB, 128] fp32
  float*       out   = (float*)d_out;          // [BATCH*HIST, 128] fp32

  const int n_rows = in_sizes[0];              // 4096 * 200 = 819200
  const int blocks = (n_rows + ROWS_PER_BLOCK - 1) / ROWS_PER_BLOCK;

  emb_gather_kernel<<<dim3(blocks), dim3(THREADS), 0, stream>>>(idx, table, out, n_rows);
}